// GATv2EdgePredictor_88356067213531
// MI455X (gfx1250) — compile-verified
//
#include <hip/hip_runtime.h>
#include <hip/hip_bf16.h>

// ---------------------------------------------------------------------------
// GATv2 edge predictor for MI455X (gfx1250, wave32).
// Dense transforms + edge MLP use V_WMMA_F32_16X16X4_F32 (full fp32 WMMA).
// Edge logits pass is wave-per-edge, exp/scatter passes thread-per-(edge,head)/(edge,8ch), with coalesced channel
// access and f32 global atomics (bandwidth-bound part of the roofline).
// ---------------------------------------------------------------------------

#define Nn 50000
#define Ee 800000
#define EDd 16
#define NEGS 0.2f

typedef float v2f __attribute__((ext_vector_type(2)));
typedef float v8f __attribute__((ext_vector_type(8)));

__device__ inline void atomicMaxF(float* addr, float value) {
  float old = *addr;
  while (old < value) {
    int assumed = __float_as_int(old);
    int prev = atomicCAS(reinterpret_cast<int*>(addr), assumed, __float_as_int(value));
    if (prev == assumed) break;
    old = __int_as_float(prev);
  }
}

// ------------------------------ utility ------------------------------------
__global__ void k_fill(float* __restrict__ p, long long n, float v) {
  long long i = blockIdx.x * (long long)blockDim.x + threadIdx.x;
  if (i < n) p[i] = v;
}

// self-loop attr: segment mean of edge_attr over dst
__global__ void k_loop_accum(const int* __restrict__ dst, const float* __restrict__ ea,
                             float* __restrict__ sum, float* __restrict__ cnt, int E) {
  int e = blockIdx.x * blockDim.x + threadIdx.x;
  if (e >= E) return;
  int d = dst[e];
  atomicAdd(&cnt[d], 1.0f);
  const float* src = ea + (long long)e * EDd;
  float* dp = sum + (long long)d * EDd;
#pragma unroll
  for (int j = 0; j < EDd; ++j) atomicAdd(&dp[j], src[j]);
}

__global__ void k_loop_div(float* __restrict__ sum, const float* __restrict__ cnt, int n16) {
  int i = blockIdx.x * blockDim.x + threadIdx.x;
  if (i >= n16) return;
  sum[i] /= fmaxf(cnt[i / EDd], 1.0f);
}

// ------------------------- WMMA fp32 GEMM (+bias) ---------------------------
// C[M x Nc] = A[M x K] @ W[K x Nc] + bias ; one wave per 16x16 tile.
__global__ void k_gemm_bias(const float* __restrict__ A, const float* __restrict__ W,
                            const float* __restrict__ bias, float* __restrict__ C,
                            int M, int K, int Nc) {
  int gtid = blockIdx.x * blockDim.x + threadIdx.x;
  int wave = gtid >> 5;
  int lane = gtid & 31;
  int tilesN = Nc >> 4;
  int tm = wave / tilesN;
  int tn = wave - tm * tilesN;
  if (tm * 16 >= M) return;           // whole-wave uniform: EXEC stays all-ones
  const int row = tm * 16 + (lane & 15);
  const int col = tn * 16 + (lane & 15);
  const int khalf = (lane >> 4) * 2;  // K sub-offset per half-wave (ISA A/B layout)
  v8f acc = {0.f, 0.f, 0.f, 0.f, 0.f, 0.f, 0.f, 0.f};
  const float* arow = A + (long long)row * K;
  for (int k = 0; k < K; k += 4) {
    v2f a, b;
    a.x = arow[k + khalf];
    a.y = arow[k + khalf + 1];
    b.x = W[(long long)(k + khalf) * Nc + col];
    b.y = W[(long long)(k + khalf + 1) * Nc + col];
    acc = __builtin_amdgcn_wmma_f32_16x16x4_f32(false, a, false, b, (short)0, acc,
                                                false, false);
  }
  float bv = bias[col];
  int mrow = tm * 16 + (lane >> 4) * 8;
#pragma unroll
  for (int r = 0; r < 8; ++r) C[(long long)(mrow + r) * Nc + col] = acc[r] + bv;
}

// -------------------- pass A: logits + segment max --------------------------
// one wave per (virtual) edge; lanes stripe channels.
__global__ void k_edge_logits(const float* __restrict__ xl, const float* __restrict__ xr,
                              const int* __restrict__ src, const int* __restrict__ dst,
                              int E, int N,
                              const float* __restrict__ edge_attr,
                              const float* __restrict__ loop_attr,
                              const float* __restrict__ We, const float* __restrict__ att,
                              float* __restrict__ logits, float* __restrict__ mx,
                              int H, int C) {
  int gtid = blockIdx.x * blockDim.x + threadIdx.x;
  int e = gtid >> 5;
  int lane = gtid & 31;
  int Et = E + N;
  if (e >= Et) return;
  int s, d;
  const float* ea;
  if (e < E) { s = src[e]; d = dst[e]; ea = edge_attr + (long long)e * EDd; }
  else       { s = d = e - E;          ea = loop_attr + (long long)(e - E) * EDd; }
  float eav = (lane < EDd) ? ea[lane] : 0.0f;
  int HC = H * C;
  float accH[2] = {0.f, 0.f};
  for (int i = 0; i < HC / 32; ++i) {
    int c = lane + 32 * i;
    float eac = 0.f;
#pragma unroll
    for (int j = 0; j < EDd; ++j) eac += __shfl(eav, j, 32) * We[j * HC + c];
    float m = xl[(long long)s * HC + c] + xr[(long long)d * HC + c] + eac;
    m = (m > 0.f) ? m : NEGS * m;     // LeakyReLU(0.2)
    accH[c / C] += m * att[c];
  }
#pragma unroll
  for (int o = 1; o < 32; o <<= 1) {
    accH[0] += __shfl_xor(accH[0], o, 32);
    accH[1] += __shfl_xor(accH[1], o, 32);
  }
  if (lane == 0) {
    for (int h = 0; h < H; ++h) {
      logits[(long long)e * H + h] = accH[h];
      atomicMaxF(&mx[d * H + h], accH[h]);
    }
  }
}

// -------------------- pass B: exp + segment sum -----------------------------
__global__ void k_edge_exp(const int* __restrict__ dst, int E, int N, int H,
                           float* __restrict__ logits, const float* __restrict__ mx,
                           float* __restrict__ den) {
  long long idx = blockIdx.x * (long long)blockDim.x + threadIdx.x;
  long long tot = (long long)(E + N) * H;
  if (idx >= tot) return;
  int e = (int)(idx / H);
  int h = (int)(idx - (long long)e * H);
  int d = (e < E) ? dst[e] : (e - E);
  float ex = expf(logits[idx] - mx[d * H + h]);
  logits[idx] = ex;                   // reuse logits buffer for exp values
  atomicAdd(&den[d * H + h], ex);
}

// -------------------- pass C: weighted scatter-add --------------------------
__global__ void k_edge_scatter(const int* __restrict__ src, const int* __restrict__ dst,
                               int E, int N, int H, int C,
                               const float* __restrict__ ex, const float* __restrict__ den,
                               const float* __restrict__ xl, float* __restrict__ hacc) {
  const int CH = 8;
  int HC = H * C;
  int chunks = HC / CH;
  long long idx = blockIdx.x * (long long)blockDim.x + threadIdx.x;
  long long tot = (long long)(E + N) * chunks;
  if (idx >= tot) return;
  int e = (int)(idx / chunks);
  int q = (int)(idx - (long long)e * chunks);
  int s, d;
  if (e < E) { s = src[e]; d = dst[e]; } else { s = d = e - E; }
  int c0 = q * CH;
  int h = c0 / C;
  float alpha = ex[(long long)e * H + h] / den[d * H + h];
  const float* xs = xl + (long long)s * HC + c0;
  float* hd = hacc + (long long)d * HC + c0;
#pragma unroll
  for (int c = 0; c < CH; ++c) atomicAdd(&hd[c], alpha * xs[c]);
}

// -------------------- node epilogue: +bias (+ELU) ---------------------------
__global__ void k_epilogue(float* __restrict__ acc, const float* __restrict__ bias,
                           int n, int HC, int do_elu) {
  long long i = blockIdx.x * (long long)blockDim.x + threadIdx.x;
  if (i >= (long long)n * HC) return;
  float v = acc[i] + bias[i % HC];
  if (do_elu) v = (v > 0.f) ? v : (expf(v) - 1.f);
  acc[i] = v;
}

// -------------------- edge MLP: WMMA 16 edges/wave --------------------------
// z = relu(concat(h[s],h[d]) @ Wm1 + bm1); out = z @ Wm2 + bm2
__global__ void k_edge_mlp(const float* __restrict__ h, const int* __restrict__ src,
                           const int* __restrict__ dst,
                           const float* __restrict__ Wm1, const float* __restrict__ bm1,
                           const float* __restrict__ Wm2, const float* __restrict__ bm2,
                           float* __restrict__ out, int E) {
  int gtid = blockIdx.x * blockDim.x + threadIdx.x;
  int wave = gtid >> 5;
  int lane = gtid & 31;
  int e0 = wave * 16;
  if (e0 >= E) return;                // uniform per wave
  int myrow = e0 + (lane & 15);
  int s = src[myrow], d = dst[myrow];
  const int khalf = (lane >> 4) * 2;
  v8f zero = {0.f, 0.f, 0.f, 0.f, 0.f, 0.f, 0.f, 0.f};
  v8f acc0 = zero, acc1 = zero, acc2 = zero, acc3 = zero;
  const int cc = lane & 15;
  for (int k = 0; k < 128; k += 4) {
    int kk = k + khalf;
    const float* arow = (kk < 64) ? (h + (long long)s * 64 + kk)
                                  : (h + (long long)d * 64 + (kk - 64));
    v2f a;
    a.x = arow[0];
    a.y = arow[1];
    const float* w0 = Wm1 + (long long)kk * 64;
    const float* w1 = Wm1 + (long long)(kk + 1) * 64;
    v2f b;
    b.x = w0[cc +  0]; b.y = w1[cc +  0];
    acc0 = __builtin_amdgcn_wmma_f32_16x16x4_f32(false, a, false, b, (short)0, acc0, false, false);
    b.x = w0[cc + 16]; b.y = w1[cc + 16];
    acc1 = __builtin_amdgcn_wmma_f32_16x16x4_f32(false, a, false, b, (short)0, acc1, false, false);
    b.x = w0[cc + 32]; b.y = w1[cc + 32];
    acc2 = __builtin_amdgcn_wmma_f32_16x16x4_f32(false, a, false, b, (short)0, acc2, false, false);
    b.x = w0[cc + 48]; b.y = w1[cc + 48];
    acc3 = __builtin_amdgcn_wmma_f32_16x16x4_f32(false, a, false, b, (short)0, acc3, false, false);
  }
  // second layer: per-edge dot(relu(z + bm1), Wm2) reduced within half-wave
  float b0 = bm1[cc +  0], b1 = bm1[cc + 16], b2 = bm1[cc + 32], b3 = bm1[cc + 48];
  float w0 = Wm2[cc +  0], w1 = Wm2[cc + 16], w2 = Wm2[cc + 32], w3 = Wm2[cc + 48];
  float bout = bm2[0];
#pragma unroll
  for (int r = 0; r < 8; ++r) {
    float z = 0.f, v;
    v = acc0[r] + b0; z += (v > 0.f ? v : 0.f) * w0;
    v = acc1[r] + b1; z += (v > 0.f ? v : 0.f) * w1;
    v = acc2[r] + b2; z += (v > 0.f ? v : 0.f) * w2;
    v = acc3[r] + b3; z += (v > 0.f ? v : 0.f) * w3;
#pragma unroll
    for (int o = 1; o < 16; o <<= 1) z += __shfl_xor(z, o, 32);  // reduce within half
    if (cc == 0) out[e0 + r + 8 * (lane >> 4)] = z + bout;
  }
}

// ---------------------------------------------------------------------------
extern "C" void kernel_launch(void* const* d_in, const int* in_sizes, int n_in,
                              void* d_out, int out_size, void* d_ws, size_t ws_size,
                              hipStream_t stream) {
  (void)in_sizes; (void)n_in; (void)out_size; (void)ws_size;
  const float* x    = (const float*)d_in[0];
  const int*   ei   = (const int*)d_in[1];
  const float* ea   = (const float*)d_in[2];
  const float* Wl1  = (const float*)d_in[3];
  const float* bl1  = (const float*)d_in[4];
  const float* Wr1  = (const float*)d_in[5];
  const float* br1  = (const float*)d_in[6];
  const float* We1  = (const float*)d_in[7];
  const float* att1 = (const float*)d_in[8];
  const float* b1   = (const float*)d_in[9];
  const float* Wl2  = (const float*)d_in[10];
  const float* bl2  = (const float*)d_in[11];
  const float* Wr2  = (const float*)d_in[12];
  const float* br2  = (const float*)d_in[13];
  const float* We2  = (const float*)d_in[14];
  const float* att2 = (const float*)d_in[15];
  const float* b2   = (const float*)d_in[16];
  const float* Wm1  = (const float*)d_in[17];
  const float* bm1  = (const float*)d_in[18];
  const float* Wm2  = (const float*)d_in[19];
  const float* bm2  = (const float*)d_in[20];
  const int* srcI = ei;            // edge_index[0]
  const int* dstI = ei + Ee;       // edge_index[1]
  float* out = (float*)d_out;

  const int Et = Nn + Ee;
  float* ws = (float*)d_ws;
  size_t o = 0;
  float* loop_attr = ws + o; o += (size_t)Nn * EDd;   // self-loop attrs (mean)
  float* cnt       = ws + o; o += (size_t)Nn;
  float* logits    = ws + o; o += (size_t)Et * 2;     // logits / exp values
  float* mx        = ws + o; o += (size_t)Nn * 2;
  float* den       = ws + o; o += (size_t)Nn * 2;
  float* xl        = ws + o; o += (size_t)Nn * 128;   // xl1, then xl2
  float* xr        = ws + o; o += (size_t)Nn * 128;   // xr1, then xr2
  float* h1        = ws + o; o += (size_t)Nn * 128;   // layer1 acc -> h1 in place
  float* h2        = ws + o; o += (size_t)Nn * 64;    // layer2 acc -> h2 in place

  const int B = 256;
  auto blk = [](long long t) { return (unsigned)((t + 255) / 256); };

  // self-loop edge attrs (segment mean over dst)
  k_fill<<<blk((long long)Nn * EDd), B, 0, stream>>>(loop_attr, (long long)Nn * EDd, 0.f);
  k_fill<<<blk(Nn), B, 0, stream>>>(cnt, Nn, 0.f);
  k_loop_accum<<<blk(Ee), B, 0, stream>>>(dstI, ea, loop_attr, cnt, Ee);
  k_loop_div<<<blk((long long)Nn * EDd), B, 0, stream>>>(loop_attr, cnt, Nn * EDd);

  // ---- layer 1 (H=2, C=64) ----
  {
    long long waves = (long long)(Nn / 16) * (128 / 16);
    k_gemm_bias<<<blk(waves * 32), B, 0, stream>>>(x, Wl1, bl1, xl, Nn, 128, 128);
    k_gemm_bias<<<blk(waves * 32), B, 0, stream>>>(x, Wr1, br1, xr, Nn, 128, 128);
  }
  k_fill<<<blk((long long)Nn * 2), B, 0, stream>>>(mx, (long long)Nn * 2, -3.0e38f);
  k_fill<<<blk((long long)Nn * 2), B, 0, stream>>>(den, (long long)Nn * 2, 0.f);
  k_fill<<<blk((long long)Nn * 128), B, 0, stream>>>(h1, (long long)Nn * 128, 0.f);
  k_edge_logits<<<blk((long long)Et * 32), B, 0, stream>>>(
      xl, xr, srcI, dstI, Ee, Nn, ea, loop_attr, We1, att1, logits, mx, 2, 64);
  k_edge_exp<<<blk((long long)Et * 2), B, 0, stream>>>(dstI, Ee, Nn, 2, logits, mx, den);
  k_edge_scatter<<<blk((long long)Et * 16), B, 0, stream>>>(
      srcI, dstI, Ee, Nn, 2, 64, logits, den, xl, h1);
  k_epilogue<<<blk((long long)Nn * 128), B, 0, stream>>>(h1, b1, Nn, 128, 1);

  // ---- layer 2 (H=1, C=64) ----
  {
    long long waves = (long long)(Nn / 16) * (64 / 16);
    k_gemm_bias<<<blk(waves * 32), B, 0, stream>>>(h1, Wl2, bl2, xl, Nn, 128, 64);
    k_gemm_bias<<<blk(waves * 32), B, 0, stream>>>(h1, Wr2, br2, xr, Nn, 128, 64);
  }
  k_fill<<<blk(Nn), B, 0, stream>>>(mx, Nn, -3.0e38f);
  k_fill<<<blk(Nn), B, 0, stream>>>(den, Nn, 0.f);
  k_fill<<<blk((long long)Nn * 64), B, 0, stream>>>(h2, (long long)Nn * 64, 0.f);
  k_edge_logits<<<blk((long long)Et * 32), B, 0, stream>>>(
      xl, xr, srcI, dstI, Ee, Nn, ea, loop_attr, We2, att2, logits, mx, 1, 64);
  k_edge_exp<<<blk(Et), B, 0, stream>>>(dstI, Ee, Nn, 1, logits, mx, den);
  k_edge_scatter<<<blk((long long)Et * 8), B, 0, stream>>>(
      srcI, dstI, Ee, Nn, 1, 64, logits, den, xl, h2);
  k_epilogue<<<blk((long long)Nn * 64), B, 0, stream>>>(h2, b2, Nn, 64, 0);

  // ---- edge MLP over original edges ----
  {
    long long waves = Ee / 16;
    k_edge_mlp<<<blk(waves * 32), B, 0, stream>>>(h2, srcI, dstI, Wm1, bm1, Wm2, bm2,
                                                  out, Ee);
  }
}